// LengthRegulator_46368466928002
// MI455X (gfx1250) — compile-verified
//
#include <hip/hip_runtime.h>

// LengthRegulator for MI455X (gfx1250, wave32).
// Zero-FLOP gather/stream: HBM floor ~ (256MB out + 32MB x) / 23.3TB/s ~ 12us.
// CDNA5 paths used:
//  - GLOBAL_LOAD_ASYNC_TO_LDS_B32 + s_wait_asynccnt (async global->LDS stage)
//  - global_prefetch_b8 (warm all 32MB of x into the 192MB L2 before gather)
//  - global_load_b128 / global_store_b128 nt (streaming output, no L2 pollution)
//  - wave32 split barriers + split wait counters (compiler-generated)

typedef float v4f __attribute__((ext_vector_type(4)));

#define B_ 32
#define T_ 512
#define D_ 512
#define L_ 4096   // max_len

// ---------------------------------------------------------------------------
// Kernel 1: per-batch duration scan + index scatter + x L2-warm.
// One block of 512 threads (16 wave32s) per batch (32 blocks total -> cheap).
// ---------------------------------------------------------------------------
__global__ __launch_bounds__(T_)
void lr_scan_scatter_kernel(const float* __restrict__ x,
                            const int* __restrict__ dur,
                            int* __restrict__ idx_ws,
                            int* __restrict__ mel_out) {
    __shared__ int s[T_];   // only LDS object in this kernel -> LDS offset 0
    const int b = blockIdx.x;
    const int t = threadIdx.x;

    // --- Stage duration row into LDS via async global->LDS copy (CDNA5). ---
    // Per-lane: one dword from the global address to the lane's LDS offset; tracked by ASYNCcnt.
    {
        unsigned           lds_off = (unsigned)(t * 4);                 // &s[t]
        unsigned long long gaddr   = (unsigned long long)(dur + (size_t)b * T_ + t);
        asm volatile("global_load_async_to_lds_b32 %0, %1, off"
                     :: "v"(lds_off), "v"(gaddr) : "memory");
        asm volatile("s_wait_asynccnt 0x0" ::: "memory");
    }
    __syncthreads();        // make all lanes' async LDS writes block-visible

    const int d = s[t];

    // --- Inclusive Hillis-Steele scan over 512 elements (9 ds steps). ---
    #pragma unroll
    for (int off = 1; off < T_; off <<= 1) {
        int v = (t >= off) ? s[t - off] : 0;
        __syncthreads();
        s[t] += v;
        __syncthreads();
    }

    const int end   = s[t];          // csum[t]
    const int start = end - d;       // csum[t-1]
    const int mel   = s[T_ - 1];     // total frames for this batch

    int* __restrict__ row = idx_ws + (size_t)b * L_;

    // Scatter: frame f in [start, end) belongs to phoneme t (durations < 8).
    for (int f = start; f < end && f < L_; ++f) row[f] = t;

    // Padded region -> -1 (gather emits zeros there). Disjoint from scatter.
    for (int f = mel + t; f < L_; f += T_) row[f] = -1;

    if (t == 0) mel_out[b] = mel;

    // --- Warm this thread's x row (2KB = 16 x 128B lines) into L2. ---
    // x (32MB) then stays L2-resident for the gather kernel's ~3.5x reuse,
    // separating read traffic from the 256MB streaming write.
    {
        const char* p = (const char*)(x + ((size_t)b * T_ + t) * (size_t)D_);
        #pragma unroll
        for (int k = 0; k < 16; ++k)
            __builtin_prefetch(p + k * 128, 0, 1);   // -> global_prefetch_b8
    }
}

// ---------------------------------------------------------------------------
// Kernel 2: one thread per output float4 (16B). Lanes of a wave cover 512
// contiguous output bytes -> perfect coalescing for both the x b128 load
// (L2 hit) and the non-temporal b128 store.
// ---------------------------------------------------------------------------
__global__ __launch_bounds__(256)
void lr_gather_kernel(const float* __restrict__ x,
                      const int* __restrict__ idx_ws,
                      float* __restrict__ out) {
    const size_t tid = (size_t)blockIdx.x * 256u + threadIdx.x; // float4 id
    const int    c4  = (int)(tid & (D_ / 4 - 1));               // chunk in row
    const size_t fj  = tid >> 7;                                // = b*L + j
    const int    idx = idx_ws[fj];

    v4f v = {0.0f, 0.0f, 0.0f, 0.0f};
    if (idx >= 0) {
        const size_t b = fj >> 12;  // L = 4096
        const v4f* __restrict__ src =
            (const v4f*)(x + ((b * T_ + (size_t)idx) * (size_t)D_)) + c4;
        v = *src;                   // global_load_b128 (L2-resident)
    }
    // Streaming output: non-temporal b128 store (TH=NT).
    __builtin_nontemporal_store(v, (v4f*)out + tid);
}

// ---------------------------------------------------------------------------
extern "C" void kernel_launch(void* const* d_in, const int* in_sizes, int n_in,
                              void* d_out, int out_size, void* d_ws, size_t ws_size,
                              hipStream_t stream) {
    (void)in_sizes; (void)n_in; (void)out_size; (void)ws_size;

    const float* x   = (const float*)d_in[0];   // [B, T, D] fp32
    const int*   dur = (const int*)d_in[1];     // [B, T] int
    // d_in[2] = max_len scalar (known: 4096)

    float* out    = (float*)d_out;                                               // [B, L, D]
    int*   mel    = (int*)((char*)d_out + (size_t)B_ * L_ * D_ * sizeof(float)); // [B]
    int*   idx_ws = (int*)d_ws;                                                  // [B, L]

    // 1) scan + scatter + L2 warm: one block per batch.
    lr_scan_scatter_kernel<<<B_, T_, 0, stream>>>(x, dur, idx_ws, mel);

    // 2) gather/stream: one thread per 16B of output.
    const size_t total4 = (size_t)B_ * L_ * (D_ / 4);   // 16,777,216
    lr_gather_kernel<<<(unsigned)(total4 / 256), 256, 0, stream>>>(x, idx_ws, out);
}